// GIN_72507637891552
// MI455X (gfx1250) — compile-verified
//
#include <hip/hip_runtime.h>
#include <cmath>

typedef __attribute__((ext_vector_type(2))) float v2f;
typedef __attribute__((ext_vector_type(4))) float v4f;
typedef __attribute__((ext_vector_type(8))) float v8f;

// ---------------------------------------------------------------------------
// Zero-fill (graph-capture-safe), vectorized.
// ---------------------------------------------------------------------------
__global__ void __launch_bounds__(256) zero_kernel(v4f* __restrict__ p, int n4) {
    int i = blockIdx.x * blockDim.x + threadIdx.x;
    if (i < n4) p[i] = (v4f){0.f, 0.f, 0.f, 0.f};
}

// ---------------------------------------------------------------------------
// Stage W [K x NC] into LDS pre-swizzled into WMMA-B fragment order:
//   sWs[((k0/4)*NT + c)*64 + lane*2 + h] = W[(k0 + 2*(lane>>4) + h)*NC + c*16 + (lane&15)]
// -> each lane's (b.x,b.y) pair is one contiguous ds_load_b64, no fixups.
// ---------------------------------------------------------------------------
template<int K, int NC>
__device__ __forceinline__ void fill_swizzled(float* sWs, const float* __restrict__ W) {
    constexpr int NT  = NC / 16;
    constexpr int TOT = (K / 4) * NT * 64;   // == K*NC
    for (int i = threadIdx.x; i < TOT; i += blockDim.x) {
        const int h    = i & 1;
        const int lane = (i >> 1) & 31;
        const int c    = (i >> 6) % NT;
        const int ks   = i / (64 * NT);
        const int k    = ks * 4 + ((lane >> 4) << 1) + h;
        const int col  = c * 16 + (lane & 15);
        sWs[i] = W[k * NC + col];
    }
}

// ---------------------------------------------------------------------------
// Fused WMMA edge kernel: per 16-edge tile,
//   LIN[16 x D] = edge_attr[16 x 16] @ eW[16 x D] + eb   (4 x NT wmma ops)
//   msg = relu(x[src] + LIN) ; atomicAdd into agg[dst]
// All WMMAs issued first, then a gather/scatter epilogue using one 32-bit
// VGPR base offset per row (col-tile deltas fold into immediate offsets).
// ---------------------------------------------------------------------------
template<int D>
__global__ void __launch_bounds__(256) edge_kernel(
    const float* __restrict__ X, const int* __restrict__ src,
    const int* __restrict__ dst, const float* __restrict__ EA,
    const float* __restrict__ eW, const float* __restrict__ eb,
    float* __restrict__ Agg, int nedges)
{
    constexpr int NT = D / 16;
    __shared__ float sWs[4 * NT * 64];       // K=16 -> 4 k-steps
    __shared__ float sB[D];
    fill_swizzled<16, D>(sWs, eW);
    for (int i = threadIdx.x; i < D; i += blockDim.x) sB[i] = eb[i];
    __syncthreads();

    const int lane  = threadIdx.x & 31;
    const int m     = lane & 15;             // A row (edge) within tile
    const int kb    = (lane >> 4) << 1;      // K-pair base: 0 or 2
    const int n     = lane & 15;             // C/D column within tile
    const int mbase = (lane >> 4) << 3;      // C/D row base: 0 or 8

    const int wavesTotal = (gridDim.x * blockDim.x) >> 5;
    const int ntiles = nedges >> 4;          // E % 16 == 0
    int wtile = (blockIdx.x * blockDim.x + threadIdx.x) >> 5;

    for (int t = wtile; t < ntiles; t += wavesTotal) {
        const int e0 = t << 4;

        // A fragments: edge_attr rows, 4 k-steps of K=4
        const int ab = (e0 + m) * 16 + kb;   // 32-bit element offset
        const v2f a0 = *(const v2f*)(EA + ab);
        const v2f a1 = *(const v2f*)(EA + ab + 4);
        const v2f a2 = *(const v2f*)(EA + ab + 8);
        const v2f a3 = *(const v2f*)(EA + ab + 12);

        // Per-row 32-bit base offsets for gather (X) and scatter (Agg)
        int sb[8], db[8];
#pragma unroll
        for (int j = 0; j < 8; ++j) {
            sb[j] = src[e0 + mbase + j] * D + n;
            db[j] = dst[e0 + mbase + j] * D + n;
        }

        // All WMMAs up front (bias folded into accumulator init)
        v8f acc[NT];
#pragma unroll
        for (int c = 0; c < NT; ++c) {
            const float bv = sB[c * 16 + n];
            acc[c] = (v8f){bv, bv, bv, bv, bv, bv, bv, bv};
            const v2f* b0 = (const v2f*)(sWs + (0 * NT + c) * 64);
            const v2f* b1 = (const v2f*)(sWs + (1 * NT + c) * 64);
            const v2f* b2 = (const v2f*)(sWs + (2 * NT + c) * 64);
            const v2f* b3 = (const v2f*)(sWs + (3 * NT + c) * 64);
            acc[c] = __builtin_amdgcn_wmma_f32_16x16x4_f32(false, a0, false, b0[lane], (short)0, acc[c], false, false);
            acc[c] = __builtin_amdgcn_wmma_f32_16x16x4_f32(false, a1, false, b1[lane], (short)0, acc[c], false, false);
            acc[c] = __builtin_amdgcn_wmma_f32_16x16x4_f32(false, a2, false, b2[lane], (short)0, acc[c], false, false);
            acc[c] = __builtin_amdgcn_wmma_f32_16x16x4_f32(false, a3, false, b3[lane], (short)0, acc[c], false, false);
        }

        // Gather + ReLU + scatter: j-outer so the NT gathers per row share one
        // VGPR offset (sb[j]) with immediate col-tile deltas, then NT atomics.
#pragma unroll
        for (int j = 0; j < 8; ++j) {
            float xv[NT];
#pragma unroll
            for (int c = 0; c < NT; ++c) xv[c] = X[sb[j] + c * 16];
#pragma unroll
            for (int c = 0; c < NT; ++c) {
                float msg = acc[c][j] + xv[c];
                msg = msg > 0.0f ? msg : 0.0f;            // ReLU
                atomicAdd(&Agg[db[j] + c * 16], msg);     // global_atomic_add_f32
            }
        }
    }
}

// ---------------------------------------------------------------------------
// WMMA f32 GEMM:  Out[N, NC] = act( (A (+A2)) [N, K] @ W [K, NC] + bias )
// One wave computes a 16-row x NC strip with V_WMMA_F32_16X16X4_F32.
// 32-bit base offsets; K-step / store deltas fold into immediate offsets.
// ---------------------------------------------------------------------------
template<int K, int NC, bool ACT_ELU, bool ADD2>
__global__ void __launch_bounds__(256) gemm_kernel(
    const float* __restrict__ A, const float* __restrict__ A2,
    const float* __restrict__ W, const float* __restrict__ bias,
    float* __restrict__ Out, int nrows)
{
    constexpr int NT = NC / 16;
    __shared__ float sWs[(K / 4) * NT * 64];
    __shared__ float sB[NC];
    fill_swizzled<K, NC>(sWs, W);
    for (int i = threadIdx.x; i < NC; i += blockDim.x) sB[i] = bias[i];
    __syncthreads();

    const int lane = threadIdx.x & 31;
    const int wave = threadIdx.x >> 5;
    const int tile = blockIdx.x * 8 + wave;      // wave-uniform
    if (tile * 16 >= nrows) return;              // uniform branch, EXEC stays all-1s
    const int row0 = tile * 16;

    const int m  = lane & 15;
    const int kb = (lane >> 4) << 1;
    const int n  = lane & 15;

    v8f acc[NT] = {};

    const int abase = (row0 + m) * K + kb;       // 32-bit element offset

#pragma unroll
    for (int ks = 0; ks < K / 4; ++ks) {
        v2f a = *(const v2f*)(A + abase + ks * 4);
        if (ADD2) {
            v2f a2 = *(const v2f*)(A2 + abase + ks * 4);
            a.x += a2.x; a.y += a2.y;            // v_pk_add_f32
        }
        const v2f* brow = (const v2f*)(sWs + ks * NT * 64);
#pragma unroll
        for (int c = 0; c < NT; ++c) {
            acc[c] = __builtin_amdgcn_wmma_f32_16x16x4_f32(
                false, a, false, brow[c * 32 + lane], (short)0, acc[c], false, false);
        }
    }

    // C/D layout: VGPR j -> row m = j + 8*(lane>>4), col n = lane&15
    const int mbase = (lane >> 4) << 3;
    const int obase = (row0 + mbase) * NC + n;   // one VGPR offset; rest immediate
#pragma unroll
    for (int c = 0; c < NT; ++c) {
        const float bv = sB[c * 16 + n];
#pragma unroll
        for (int j = 0; j < 8; ++j) {
            float v = acc[c][j] + bv;
            if (ACT_ELU) v = (v > 0.0f) ? v : (__expf(v) - 1.0f);   // ELU, native exp
            Out[obase + j * NC + c * 16] = v;
        }
    }
}

// ---------------------------------------------------------------------------
// Host-side launch. Inputs in setup_inputs() order.
// ---------------------------------------------------------------------------
extern "C" void kernel_launch(void* const* d_in, const int* in_sizes, int n_in,
                              void* d_out, int out_size, void* d_ws, size_t ws_size,
                              hipStream_t stream)
{
    const float* x    = (const float*)d_in[0];
    const int*   src  = (const int*)  d_in[1];
    const int*   dst  = (const int*)  d_in[2];
    const float* ea   = (const float*)d_in[3];
    const float* eW0  = (const float*)d_in[4];
    const float* eb0  = (const float*)d_in[5];
    const float* W1_0 = (const float*)d_in[6];
    const float* b1_0 = (const float*)d_in[7];
    const float* W2_0 = (const float*)d_in[8];
    const float* b2_0 = (const float*)d_in[9];
    const float* eW_r = (const float*)d_in[10];
    const float* eb_r = (const float*)d_in[11];
    const float* W1_r = (const float*)d_in[12];
    const float* b1_r = (const float*)d_in[13];
    const float* W2_r = (const float*)d_in[14];
    const float* b2_r = (const float*)d_in[15];
    const float* cW   = (const float*)d_in[16];
    const float* cb   = (const float*)d_in[17];

    const int N = in_sizes[0] / 64;   // 50000
    const int E = in_sizes[1];        // 800000

    // Workspace: agg[N,128] | h[N,128] | t[N,128]  (~77 MB)
    float* agg = (float*)d_ws;
    float* hb  = agg + (size_t)N * 128;
    float* tb  = hb  + (size_t)N * 128;

    const int tiles   = (N + 15) / 16;        // 3125 (exact)
    const int gblocks = (tiles + 7) / 8;      // 8 waves/block
    const int eblocks = 2048;                 // 16384 waves over 50000 edge tiles

    // ---- Layer 0 (D=64) ----
    { int n4 = (N * 64) / 4; zero_kernel<<<(n4 + 255) / 256, 256, 0, stream>>>((v4f*)agg, n4); }
    edge_kernel<64><<<eblocks, 256, 0, stream>>>(x, src, dst, ea, eW0, eb0, agg, E);
    gemm_kernel<64, 64, true, true ><<<gblocks, 256, 0, stream>>>(x,  agg, W1_0, b1_0, tb, N);
    gemm_kernel<64, 128, true, false><<<gblocks, 256, 0, stream>>>(tb, tb,  W2_0, b2_0, hb, N);

    // ---- Layers 1..2 (D=128) ----
    for (int i = 0; i < 2; ++i) {
        int n4 = (N * 128) / 4;
        zero_kernel<<<(n4 + 255) / 256, 256, 0, stream>>>((v4f*)agg, n4);
        edge_kernel<128><<<eblocks, 256, 0, stream>>>(
            hb, src, dst, ea, eW_r + (size_t)i * 16 * 128, eb_r + (size_t)i * 128, agg, E);
        gemm_kernel<128, 128, true, true ><<<gblocks, 256, 0, stream>>>(
            hb, agg, W1_r + (size_t)i * 128 * 128, b1_r + (size_t)i * 128, tb, N);
        gemm_kernel<128, 128, true, false><<<gblocks, 256, 0, stream>>>(
            tb, tb, W2_r + (size_t)i * 128 * 128, b2_r + (size_t)i * 128, hb, N);
    }

    // ---- Classifier: out = h @ cW + cb  [N, 32] ----
    gemm_kernel<128, 32, false, false><<<gblocks, 256, 0, stream>>>(
        hb, hb, cW, cb, (float*)d_out, N);
}